// BatchSampler_87368224735453
// MI455X (gfx1250) — compile-verified
//
#include <hip/hip_runtime.h>
#include <hip/hip_bf16.h>
#include <stdint.h>

// ---------------- problem constants (from reference setup_inputs) ----------
#define N_NODES 10000
#define N_EDGES 5000000
#define BATCH   2000
#define NPAIRS  1999000               // BATCH*(BATCH-1)/2
static_assert((long long)BATCH * (BATCH - 1) / 2 == NPAIRS, "pairs");

#define RADIX_BITS 9
#define BINS       512
#define CHUNK      4096
#define NBLK       ((N_EDGES + CHUNK - 1) / CHUNK)   // 1221

// ---------------- output layout (tuple concatenated, float32) --------------
#define OFF_NODES  0
#define OFF_PAIRS  (OFF_NODES + BATCH)            // 2000
#define OFF_EDGES  (OFF_PAIRS + 2 * NPAIRS)       // 4,000,000
#define OFF_TIMES  (OFF_EDGES + 2 * N_EDGES)      // 14,000,000
#define OFF_STATES (OFF_TIMES + N_EDGES)          // 19,000,000
#define OFF_MASK   (OFF_STATES + N_EDGES)         // 24,000,000

typedef __attribute__((ext_vector_type(2))) float v2f;
typedef __attribute__((ext_vector_type(8))) float v8f;

#ifndef __has_builtin
#define __has_builtin(x) 0
#endif
#if __has_builtin(__builtin_amdgcn_wmma_f32_16x16x4_f32)
#define HAVE_WMMA_F32X4 1
#else
#define HAVE_WMMA_F32X4 0
#endif

// ---------------- CDNA5 async-to-LDS helpers (ASYNCcnt path) ---------------
__device__ __forceinline__ void async_copy_b128(void* lds_dst, const void* gsrc) {
#if defined(__gfx1250__)
  // Generic LDS pointer: addr[31:0] is the LDS byte address (ISA 10.2).
  uint32_t lds_off = (uint32_t)(uintptr_t)lds_dst;
  asm volatile("global_load_async_to_lds_b128 %0, %1, off"
               :: "v"(lds_off), "v"(gsrc) : "memory");
#else
  // host / non-gfx1250 fallback (never executed on device)
  *(uint4*)lds_dst = *(const uint4*)gsrc;
#endif
}
__device__ __forceinline__ void async_wait0() {
#if defined(__gfx1250__)
  asm volatile("s_wait_asynccnt 0" ::: "memory");
#endif
}

// Stage a CHUNK-sized u32 tile into LDS (full tiles via async b128 copies).
__device__ __forceinline__ void stage_tile(uint32_t* tile, const uint32_t* gsrc,
                                           int count, int t) {
  if (count == CHUNK) {
    for (int i = t * 4; i < CHUNK; i += 256 * 4)
      async_copy_b128(&tile[i], &gsrc[i]);
    async_wait0();
  } else {
    for (int i = t; i < CHUNK; i += 256)
      tile[i] = (i < count) ? gsrc[i] : 0u;
  }
  __syncthreads();
}

// ---------------- threefry2x32 (exact JAX PRNG, key = (0,19)) --------------
__device__ __forceinline__ uint32_t rotl32(uint32_t x, int d) {
  return (x << d) | (x >> (32 - d));
}
__device__ __forceinline__ float score_from_bits(uint32_t bits, int i) {
  const float tiny = 1.17549435e-38f;                 // float32 tiny
  uint32_t fb = (bits >> 9) | 0x3f800000u;
  float u = __uint_as_float(fb) - 1.0f;               // [0,1)
  u = u * (1.0f - tiny) + tiny;
  u = fmaxf(u, tiny);                                  // jax uniform: max(minval, .)
  float g = -logf(-logf(u));                           // gumbel
  float lw = (i > 0) ? logf((float)i) : -__builtin_huge_valf();
  return lw + g;
}

__global__ void gumbel_scores(float* scores) {
  int p = blockIdx.x * 256 + threadIdx.x;
  const int H = N_NODES / 2;                           // 5000 (even split)
  if (p >= H) return;
  uint32_t x0 = (uint32_t)p, x1 = (uint32_t)(p + H);
  const uint32_t k0 = 0u, k1 = 19u, k2 = 0u ^ 19u ^ 0x1BD11BDAu;
  x0 += k0; x1 += k1;
  const int rotA[4] = {13, 15, 26, 6};
  const int rotB[4] = {17, 29, 16, 24};
#define TF_R4(R) for (int q = 0; q < 4; ++q) { x0 += x1; x1 = rotl32(x1, R[q]); x1 ^= x0; }
  TF_R4(rotA); x0 += k1; x1 += k2 + 1u;
  TF_R4(rotB); x0 += k2; x1 += k0 + 2u;
  TF_R4(rotA); x0 += k0; x1 += k1 + 3u;
  TF_R4(rotB); x0 += k1; x1 += k2 + 4u;
  TF_R4(rotA); x0 += k2; x1 += k0 + 5u;
#undef TF_R4
  scores[p]     = score_from_bits(x0, p);
  scores[p + H] = score_from_bits(x1, p + H);
}

// exact lax.top_k semantics: rank = #{j: s_j > s_i} + #{j < i: s_j == s_i}
__global__ void rank_select(const float* scores, uint32_t* flags) {
  __shared__ float sh[256];
  int i = blockIdx.x * 256 + threadIdx.x;
  float si = (i < N_NODES) ? scores[i] : 0.f;
  int rank = 0;
  for (int base = 0; base < N_NODES; base += 256) {
    int j = base + (int)threadIdx.x;
    sh[threadIdx.x] = (j < N_NODES) ? scores[j] : -__builtin_huge_valf();
    __syncthreads();
    int lim = min(256, N_NODES - base);
    for (int k = 0; k < lim; ++k) {
      float sj = sh[k];
      int jj = base + k;
      rank += (sj > si) || (sj == si && jj < i);
    }
    __syncthreads();
  }
  if (i < N_NODES) flags[i] = (rank < BATCH) ? 1u : 0u;
}

// compact selected node ids in ascending order (== jnp.sort(sel))
__global__ __launch_bounds__(1024) void compact_batch(const uint32_t* flags,
                                                      uint32_t* bn, float* out_nodes) {
  __shared__ uint32_t f[N_NODES];
  __shared__ uint32_t segsum[1024];
  int t = threadIdx.x;
  for (int i = t; i < N_NODES; i += 1024) f[i] = flags[i];
  __syncthreads();
  const int SEG = (N_NODES + 1023) / 1024;             // 10
  int s0 = t * SEG;
  uint32_t s = 0;
  for (int k = 0; k < SEG; ++k) { int i = s0 + k; if (i < N_NODES) s += f[i]; }
  segsum[t] = s;
  __syncthreads();
  if (t == 0) {
    uint32_t run = 0;
    for (int k = 0; k < 1024; ++k) { uint32_t v = segsum[k]; segsum[k] = run; run += v; }
  }
  __syncthreads();
  uint32_t pos = segsum[t];
  for (int k = 0; k < SEG; ++k) {
    int i = s0 + k;
    if (i < N_NODES && f[i]) {
      bn[pos] = (uint32_t)i;
      out_nodes[pos] = (float)i;
      ++pos;
    }
  }
}

// triu_indices(BATCH, k=1) pairs of batch nodes
__global__ void build_pairs(const uint32_t* bn, float* out_pairs) {
  __shared__ uint32_t b[BATCH];
  int t = threadIdx.x;
  for (int i = t; i < BATCH; i += 256) b[i] = bn[i];
  __syncthreads();
  long long p = (long long)blockIdx.x * 256 + t;
  if (p >= NPAIRS) return;
  // cum(a) = a*(2B-1-a)/2 ; largest a with cum(a) <= p
  int lo = 0, hi = BATCH - 2;
  while (lo < hi) {
    int mid = (lo + hi + 1) >> 1;
    long long c = (long long)mid * (2 * BATCH - 1 - mid) / 2;
    if (c <= p) lo = mid; else hi = mid - 1;
  }
  int a = lo;
  long long ca = (long long)a * (2 * BATCH - 1 - a) / 2;
  int bidx = a + 1 + (int)(p - ca);
  out_pairs[p]          = (float)b[a];
  out_pairs[NPAIRS + p] = (float)b[bidx];
}

// keys = triu flat index (< 2^26), vals = original edge index
__global__ void build_keys(const int* e0, const int* e1, uint32_t* keys, uint32_t* vals) {
  long long e = (long long)blockIdx.x * 256 + threadIdx.x;
  if (e >= N_EDGES) return;
  long long i = e0[e], j = e1[e];
  long long flat = i * (long long)N_NODES - i * (i + 1) / 2 + (j - i - 1);
  keys[e] = (uint32_t)flat;
  vals[e] = (uint32_t)e;
}

// ---------------- radix sort: histogram --------------------------------
__global__ __launch_bounds__(256) void radix_hist(const uint32_t* keys,
                                                  uint32_t* hist, int shift) {
  __shared__ __attribute__((aligned(16))) uint32_t tile[CHUNK];
  __shared__ uint32_t lcnt[BINS];
  int t = threadIdx.x;
  int blk = blockIdx.x;
  for (int i = t; i < BINS; i += 256) lcnt[i] = 0;
  long long base = (long long)blk * CHUNK;
  int count = (int)((base + CHUNK <= N_EDGES) ? CHUNK : (N_EDGES - base));
  stage_tile(tile, keys + base, count, t);   // barrier inside (after lcnt init: ok, same sync)
  __syncthreads();
  for (int i = t; i < count; i += 256) {
    uint32_t d = (tile[i] >> shift) & (BINS - 1);
    atomicAdd(&lcnt[d], 1u);
  }
  __syncthreads();
  for (int i = t; i < BINS; i += 256) hist[(size_t)i * NBLK + blk] = lcnt[i];
}

// per bin: exclusive scan over per-block counts (in place) + bin total
__global__ __launch_bounds__(32) void per_bin_block_scan(uint32_t* hist, uint32_t* binTotal) {
  int b = blockIdx.x;
  int lane = threadIdx.x;
  uint32_t carry = 0;
  for (int base = 0; base < NBLK; base += 32) {
    int k = base + lane;
    uint32_t v = (k < NBLK) ? hist[(size_t)b * NBLK + k] : 0u;
    uint32_t s = v;
    for (int d = 1; d < 32; d <<= 1) {
      uint32_t u = __shfl_up(s, d, 32);
      if (lane >= d) s += u;
    }
    uint32_t excl = s - v + carry;
    if (k < NBLK) hist[(size_t)b * NBLK + k] = excl;
    carry += __shfl(s, 31, 32);
  }
  if (lane == 0) binTotal[b] = carry;
}

// exclusive scan of the 512 bin totals — tensor-core scan:
//   RowIncl = M x U (upper-tri ones), RowOff = Lstrict x broadcast(rowTot)
// exact in f32 since all counts/prefixes <= 5e6 < 2^24.
__global__ __launch_bounds__(32) void binstart_scan(const uint32_t* binTotal,
                                                    uint32_t* binStart) {
#if HAVE_WMMA_F32X4
  __shared__ float xs[256];
  __shared__ float rowincl[256];
  int lane = threadIdx.x;
  int col = lane & 15;              // = M index for A frags, N index for B/C/D frags
  int hi = lane >> 4;               // lane group (0: lanes 0-15, 1: lanes 16-31)
  float carry = 0.f;
  for (int c = 0; c < BINS / 256; ++c) {
    for (int i = lane; i < 256; i += 32) xs[i] = (float)binTotal[c * 256 + i];
    __syncthreads();
    // Stage 1: RowIncl = M(16x16) x U(16x16, r<=c ones), K split in 4 chunks of 4
    v8f accR = {0.f, 0.f, 0.f, 0.f, 0.f, 0.f, 0.f, 0.f};
    for (int k = 0; k < 4; ++k) {
      int kk = 4 * k + 2 * hi;      // A layout: lanes 0-15 K={4k,4k+1}, lanes 16-31 K={4k+2,4k+3}
      v2f a, b;
      a.x = xs[col * 16 + kk];
      a.y = xs[col * 16 + kk + 1];
      b.x = (kk     <= col) ? 1.f : 0.f;   // U[kk][col]
      b.y = (kk + 1 <= col) ? 1.f : 0.f;
      accR = __builtin_amdgcn_wmma_f32_16x16x4_f32(false, a, false, b,
                                                   (short)0, accR, false, false);
    }
    for (int v = 0; v < 8; ++v) rowincl[(v + 8 * hi) * 16 + col] = accR[v];
    __syncthreads();
    float chunk_total = 0.f;
    for (int r = 0; r < 16; ++r) chunk_total += rowincl[r * 16 + 15];
    // Stage 2: RowOff = Lstrict(16x16, c<r ones) x V (rowTot broadcast along N)
    v8f accO = {0.f, 0.f, 0.f, 0.f, 0.f, 0.f, 0.f, 0.f};
    for (int k = 0; k < 4; ++k) {
      int kk = 4 * k + 2 * hi;
      v2f a, b;
      a.x = (kk     < col) ? 1.f : 0.f;    // Lstrict[col][kk]
      a.y = (kk + 1 < col) ? 1.f : 0.f;
      b.x = rowincl[kk * 16 + 15];         // rowTot[kk] broadcast along N
      b.y = rowincl[(kk + 1) * 16 + 15];
      accO = __builtin_amdgcn_wmma_f32_16x16x4_f32(false, a, false, b,
                                                   (short)0, accO, false, false);
    }
    for (int v = 0; v < 8; ++v) {
      int r = v + 8 * hi;
      float excl = accR[v] - xs[r * 16 + col] + accO[v] + carry;
      binStart[c * 256 + r * 16 + col] = (uint32_t)(excl + 0.5f);
    }
    carry += chunk_total;
    __syncthreads();
  }
#else
  if (threadIdx.x == 0) {
    uint32_t run = 0;
    for (int i = 0; i < BINS; ++i) { uint32_t v = binTotal[i]; binStart[i] = run; run += v; }
  }
#endif
}

// ---------------- radix sort: stable scatter -------------------------------
__global__ __launch_bounds__(256) void radix_scatter(const uint32_t* keysIn,
                                                     const uint32_t* valsIn,
                                                     uint32_t* keysOut, uint32_t* valsOut,
                                                     const uint32_t* hist,
                                                     const uint32_t* binStart, int shift) {
  __shared__ __attribute__((aligned(16))) uint32_t tile[CHUNK];   // 16 KB
  __shared__ unsigned short rankLds[CHUNK];                       //  8 KB
  __shared__ uint32_t cnt[8][BINS];                               // 16 KB
  __shared__ uint32_t waveExcl[8][BINS];                          // 16 KB
  int t = threadIdx.x, lane = t & 31, w = t >> 5;
  long long base = (long long)blockIdx.x * CHUNK;
  int count = (int)((base + CHUNK <= N_EDGES) ? CHUNK : (N_EDGES - base));
  for (int i = t; i < 8 * BINS; i += 256) (&cnt[0][0])[i] = 0;
  stage_tile(tile, keysIn + base, count, t);
  __syncthreads();
  // Phase 1: per-wave stable digit ranks over the wave's contiguous sub-chunk
  const int SUB = CHUNK / 8;                       // 512 elements per wave
  int sbase = w * SUB;
  for (int g = 0; g < SUB / 32; ++g) {
    int idx = sbase + g * 32 + lane;
    uint32_t key = tile[idx];
    unsigned valid = (idx < count) ? 1u : 0u;
    uint32_t d = (key >> shift) & (BINS - 1);
    uint32_t before = 0, same = 0;
    for (int j = 0; j < 32; ++j) {
      uint32_t dj = __shfl(d, j, 32);
      unsigned vj = __shfl(valid, j, 32);
      bool eq = vj && (dj == d);
      same += eq ? 1u : 0u;
      if (eq && j < lane) ++before;
    }
    uint32_t baseCnt = valid ? cnt[w][d] : 0u;     // same-wave lockstep read...
    if (valid && before == 0) cnt[w][d] = baseCnt + same;  // ...then leader write
    if (valid) rankLds[idx] = (unsigned short)(baseCnt + before);
  }
  __syncthreads();
  // Phase 2: cross-wave exclusive prefix per digit (wave order == element order)
  for (int d = t; d < BINS; d += 256) {
    uint32_t run = 0;
    for (int wv = 0; wv < 8; ++wv) { waveExcl[wv][d] = run; run += cnt[wv][d]; }
  }
  __syncthreads();
  // Phase 3: scatter (stable: bin base + block offset + wave offset + in-wave rank)
  for (int i = t; i < count; i += 256) {
    uint32_t key = tile[i];
    uint32_t d = (key >> shift) & (BINS - 1);
    int we = i / SUB;
    size_t pos = (size_t)binStart[d] + hist[(size_t)d * NBLK + blockIdx.x]
               + waveExcl[we][d] + rankLds[i];
    keysOut[pos] = key;
    valsOut[pos] = valsIn[base + i];
  }
}

// ---------------- final gather + membership mask --------------------------
__global__ void final_gather(const uint32_t* order, const int* e0, const int* e1,
                             const float* times, const int* states,
                             const uint32_t* inb, float* out) {
  long long s = (long long)blockIdx.x * 256 + threadIdx.x;
  if (s >= N_EDGES) return;
  uint32_t idx = order[s];
  int i = e0[idx], j = e1[idx];
  out[OFF_EDGES + s]           = (float)i;
  out[OFF_EDGES + N_EDGES + s] = (float)j;
  bool m = inb[i] && inb[j];   // == (batch_flat[searchsorted] == flat), since flat is bijective on i<j
  out[OFF_TIMES + s]  = m ? times[idx] : 0.f;
  out[OFF_STATES + s] = m ? (float)states[idx] : 0.f;
  out[OFF_MASK + s]   = m ? 1.f : 0.f;
}

// ---------------- host driver ---------------------------------------------
extern "C" void kernel_launch(void* const* d_in, const int* in_sizes, int n_in,
                              void* d_out, int out_size, void* d_ws, size_t ws_size,
                              hipStream_t stream) {
  (void)in_sizes; (void)n_in; (void)out_size; (void)ws_size;
  const int*   edges  = (const int*)d_in[0];    // (2, E) row-major
  const float* times  = (const float*)d_in[1];
  const int*   states = (const int*)d_in[2];
  float* out = (float*)d_out;
  char* ws = (char*)d_ws;

  const size_t KB = 1024, MB = 1024 * 1024;
  float*    scores   = (float*)   (ws + 0);          // 40 KB
  uint32_t* flags    = (uint32_t*)(ws + 64 * KB);    // 40 KB (doubles as in_batch)
  uint32_t* bn       = (uint32_t*)(ws + 128 * KB);   //  8 KB
  uint32_t* binTotal = (uint32_t*)(ws + 192 * KB);   //  2 KB
  uint32_t* binStart = (uint32_t*)(ws + 196 * KB);   //  2 KB
  uint32_t* hist     = (uint32_t*)(ws + 256 * KB);   // 512*1221*4 ~ 2.4 MB
  uint32_t* keysA    = (uint32_t*)(ws + 4 * MB);     // 20 MB
  uint32_t* valsA    = (uint32_t*)(ws + 24 * MB);    // 20 MB
  uint32_t* keysB    = (uint32_t*)(ws + 44 * MB);    // 20 MB
  uint32_t* valsB    = (uint32_t*)(ws + 64 * MB);    // 20 MB

  gumbel_scores<<<(N_NODES / 2 + 255) / 256, 256, 0, stream>>>(scores);
  rank_select<<<(N_NODES + 255) / 256, 256, 0, stream>>>(scores, flags);
  compact_batch<<<1, 1024, 0, stream>>>(flags, bn, out + OFF_NODES);
  build_pairs<<<(NPAIRS + 255) / 256, 256, 0, stream>>>(bn, out + OFF_PAIRS);
  build_keys<<<(N_EDGES + 255) / 256, 256, 0, stream>>>(edges, edges + N_EDGES,
                                                        keysA, valsA);

  const int shifts[3] = {0, RADIX_BITS, 2 * RADIX_BITS};   // 27 bits >= 26-bit keys
  uint32_t *ki = keysA, *vi = valsA, *ko = keysB, *vo = valsB;
  for (int p = 0; p < 3; ++p) {
    radix_hist<<<NBLK, 256, 0, stream>>>(ki, hist, shifts[p]);
    per_bin_block_scan<<<BINS, 32, 0, stream>>>(hist, binTotal);
    binstart_scan<<<1, 32, 0, stream>>>(binTotal, binStart);
    radix_scatter<<<NBLK, 256, 0, stream>>>(ki, vi, ko, vo, hist, binStart, shifts[p]);
    uint32_t* tk = ki; ki = ko; ko = tk;
    uint32_t* tv = vi; vi = vo; vo = tv;
  }
  // after 3 swaps the sorted permutation lives in vi (= valsB)
  final_gather<<<(N_EDGES + 255) / 256, 256, 0, stream>>>(vi, edges, edges + N_EDGES,
                                                          times, states, flags, out);
}